// PFNLayerV15_44092134261321
// MI455X (gfx1250) — compile-verified
//
#include <hip/hip_runtime.h>
#include <math.h>

typedef __attribute__((ext_vector_type(2))) float v2f;
typedef __attribute__((ext_vector_type(8))) float v8f;

#define VMAX 50000
#define IN_CH 10
#define C_OUT 32

// ---------- helpers ----------
__device__ __forceinline__ unsigned fmap(float f) {
    unsigned b = __float_as_uint(f);
    return (b & 0x80000000u) ? ~b : (b | 0x80000000u);
}
__device__ __forceinline__ float funmap(unsigned u) {
    unsigned b = (u & 0x80000000u) ? (u & 0x7FFFFFFFu) : ~u;
    return __uint_as_float(b);
}
__device__ __forceinline__ float wave_sum(float v) {
    for (int o = 16; o >= 1; o >>= 1) v += __shfl_xor(v, o, 32);
    return v;
}
__device__ __forceinline__ v8f wmma_f32_k4(v2f a, v2f b, v8f c) {
    return __builtin_amdgcn_wmma_f32_16x16x4_f32(false, a, false, b, (short)0, c, false, false);
}
__device__ __forceinline__ int iclamp(int v, int hi) { return v < hi ? v : hi; }

// ---------- kernel 0: init workspace ----------
__global__ void k_init(unsigned* qmax, unsigned* minb, float* segsum,
                       unsigned* wmaxb, float* denom, int* cnt, float* stats) {
    long i = (long)blockIdx.x * blockDim.x + threadIdx.x;
    if (i < (long)VMAX * C_OUT) {
        qmax[i] = 0u;            // x >= 0 -> raw bits ordered; empty seg decodes to 0
        minb[i] = 0xFFFFFFFFu;   // sentinel for empty -> 0
        segsum[i] = 0.f;
    }
    if (i < VMAX) { wmaxb[i] = 0u; denom[i] = 0.f; cnt[i] = 0; }
    if (i < 65) stats[i] = 0.f;
}

// ---------- kernel 1: input stats (sum_x[10], upper-tri sum_xx[55]) ----------
__global__ void k_stats(const float* __restrict__ in, float* __restrict__ st, int N) {
    long tid = (long)blockIdx.x * blockDim.x + threadIdx.x;
    long stride = (long)gridDim.x * blockDim.x;
    float acc[65];
#pragma unroll
    for (int m = 0; m < 65; ++m) acc[m] = 0.f;
    for (long r = tid; r < N; r += stride) {
        float x[IN_CH];
#pragma unroll
        for (int k = 0; k < IN_CH; ++k) x[k] = in[r * IN_CH + k];
#pragma unroll
        for (int k = 0; k < IN_CH; ++k) acc[k] += x[k];
        int idx = IN_CH;
#pragma unroll
        for (int j = 0; j < IN_CH; ++j)
#pragma unroll
            for (int k = j; k < IN_CH; ++k) acc[idx++] += x[j] * x[k];
    }
    int lane = threadIdx.x & 31;
#pragma unroll
    for (int m = 0; m < 65; ++m) {
        float v = wave_sum(acc[m]);
        if (lane == 0) atomicAdd(&st[m], v);
    }
}

// ---------- kernel 2: fold BN into per-channel affine (both linear layers) ----------
__global__ void k_bnparams(const float* __restrict__ st,
                           const float* __restrict__ Wl, const float* __restrict__ g_n,
                           const float* __restrict__ be_n,
                           const float* __restrict__ Ww1, const float* __restrict__ g_w,
                           const float* __restrict__ be_w,
                           float* __restrict__ params, int N) {
    int t = threadIdx.x;
    if (t >= 64) return;
    float invN = 1.f / (float)N;
    float mx[IN_CH];
#pragma unroll
    for (int k = 0; k < IN_CH; ++k) mx[k] = st[k] * invN;
    int layer = t >> 5;
    int c = t & 31;
    const float* W  = layer ? Ww1 : Wl;
    const float* g  = layer ? g_w : g_n;
    const float* be = layer ? be_w : be_n;
    float eps = layer ? 1e-5f : 1e-3f;
    float wc[IN_CH];
#pragma unroll
    for (int k = 0; k < IN_CH; ++k) wc[k] = W[k * C_OUT + c];
    float meanLin = 0.f;
#pragma unroll
    for (int k = 0; k < IN_CH; ++k) meanLin += mx[k] * wc[k];
    float var = 0.f;
    int idx = IN_CH;
#pragma unroll
    for (int j = 0; j < IN_CH; ++j)
#pragma unroll
        for (int k = j; k < IN_CH; ++k) {
            float cv = st[idx++] * invN - mx[j] * mx[k];
            var += ((j == k) ? 1.f : 2.f) * cv * wc[j] * wc[k];
        }
    float inv = rsqrtf(var + eps);
    float scale = g[c] * inv;
    float shift = be[c] - meanLin * scale;   // applied to raw accumulator (bias folded)
    params[layer * 64 + c] = scale;
    params[layer * 64 + 32 + c] = shift;
}

// ---------- A-fragment loader: unconditional clamped loads + cndmask zeroing ----------
__device__ __forceinline__ void load_a_frags(const float* __restrict__ in, long rowBase,
                                             int n, int half, int N, v2f a[3]) {
    long arow = rowBase + n;
    bool av = arow < N;
    long arc = av ? arow : (long)(N - 1);   // clamp: always a valid row
#pragma unroll
    for (int s = 0; s < 3; ++s) {
        int kk = 4 * s + 2 * half;
        float vx = in[arc * IN_CH + iclamp(kk, IN_CH - 1)];
        float vy = in[arc * IN_CH + iclamp(kk + 1, IN_CH - 1)];
        a[s].x = (av && kk     < IN_CH) ? vx : 0.f;
        a[s].y = (av && kk + 1 < IN_CH) ? vy : 0.f;
    }
}
// ---------- B-fragment loader (weight matrix [10,32]) ----------
__device__ __forceinline__ v2f load_b_frag(const float* __restrict__ W, int s, int half, int col) {
    int kk = 4 * s + 2 * half;
    float bx = W[iclamp(kk,     IN_CH - 1) * C_OUT + col];
    float by = W[iclamp(kk + 1, IN_CH - 1) * C_OUT + col];
    v2f b;
    b.x = (kk     < IN_CH) ? bx : 0.f;
    b.y = (kk + 1 < IN_CH) ? by : 0.f;
    return b;
}

// ---------- kernel 3: WMMA main pass over rows ----------
// per wave: 16 rows. v & h GEMMs via v_wmma_f32_16x16x4_f32 (K=10 padded to 12).
__global__ void k_main(const float* __restrict__ in, const int* __restrict__ seg,
                       const float* __restrict__ Wl, const float* __restrict__ Ww1,
                       const float* __restrict__ Ww2, const float* __restrict__ bw2,
                       const float* __restrict__ params,
                       float* __restrict__ out, unsigned* __restrict__ qmax,
                       unsigned* __restrict__ minb, unsigned* __restrict__ wmaxb,
                       float* __restrict__ weight, int N) {
    long wave = ((long)blockIdx.x * blockDim.x + threadIdx.x) >> 5;
    int lane = threadIdx.x & 31;
    int half = lane >> 4, n = lane & 15;
    long rowBase = wave * 16;
    if (rowBase >= N) return;                 // wave-uniform

    v2f a[3];
    load_a_frags(in, rowBase, n, half, N, a);

    v8f vacc[2] = {};
    v8f hacc[2] = {};
#pragma unroll
    for (int t = 0; t < 2; ++t) {
        int col = n + 16 * t;
#pragma unroll
        for (int s = 0; s < 3; ++s) {
            v2f bl = load_b_frag(Wl,  s, half, col);
            v2f bw = load_b_frag(Ww1, s, half, col);
            vacc[t] = wmma_f32_k4(a[s], bl, vacc[t]);
            hacc[t] = wmma_f32_k4(a[s], bw, hacc[t]);
        }
    }
    int seg8[8]; bool rv[8];
#pragma unroll
    for (int i = 0; i < 8; ++i) {
        long r = rowBase + i + 8 * half;
        rv[i] = (r < N);
        seg8[i] = seg[rv[i] ? r : (long)(N - 1)];   // clamped unconditional load
    }
    const float* scale_v = params;
    const float* shift_v = params + 32;
    const float* scale_h = params + 64;
    const float* shift_h = params + 96;
    float w2a = Ww2[n], w2b = Ww2[n + 16];
#pragma unroll
    for (int i = 0; i < 8; ++i) {
        long r = rowBase + i + 8 * half;
        float p = 0.f;
#pragma unroll
        for (int t = 0; t < 2; ++t) {
            int c = n + 16 * t;
            int sidx = seg8[i] * C_OUT + c;        // 32-bit index (V*32 < 2^31)
            float nv = fmaf(vacc[t][i], scale_v[c], shift_v[c]);
            float x = fmaxf(nv, 0.f);
            float hn = fmaxf(fmaf(hacc[t][i], scale_h[c], shift_h[c]), 0.f);
            p += hn * (t ? w2b : w2a);
            if (rv[i]) {
                out[r * 64 + c] = x;
                atomicMax(&qmax[sidx], __float_as_uint(x));
                atomicMin(&minb[sidx], fmap(nv));
            }
        }
        // reduce p across the 16 columns of this half
        for (int o = 8; o >= 1; o >>= 1) p += __shfl_xor(p, o, 16);
        if (n == 0 && rv[i]) {
            float w = p + bw2[0];
            weight[r] = w;
            atomicMax(&wmaxb[seg8[i]], fmap(w));
        }
    }
}

// ---------- kernel 4: softmax denominators + counts ----------
__global__ void k_denom(const int* __restrict__ seg, const float* __restrict__ weight,
                        const unsigned* __restrict__ wmaxb, float* __restrict__ denom,
                        int* __restrict__ cnt, int N) {
    long r = (long)blockIdx.x * blockDim.x + threadIdx.x;
    if (r >= N) return;
    int s = seg[r];
    float e = expf(weight[r] - funmap(wmaxb[s]));
    atomicAdd(&denom[s], e);
    atomicAdd(&cnt[s], 1);
}

// ---------- kernel 5: WMMA recompute v, scatter softmax-weighted sum ----------
__global__ void k_wsum(const float* __restrict__ in, const int* __restrict__ seg,
                       const float* __restrict__ Wl, const float* __restrict__ bl_in,
                       const float* __restrict__ weight, const unsigned* __restrict__ wmaxb,
                       const float* __restrict__ denom, float* __restrict__ segsum, int N) {
    long wave = ((long)blockIdx.x * blockDim.x + threadIdx.x) >> 5;
    int lane = threadIdx.x & 31;
    int half = lane >> 4, n = lane & 15;
    long rowBase = wave * 16;
    if (rowBase >= N) return;

    v2f a[3];
    load_a_frags(in, rowBase, n, half, N, a);

    v8f vacc[2] = {};
#pragma unroll
    for (int t = 0; t < 2; ++t) {
        int col = n + 16 * t;
#pragma unroll
        for (int s = 0; s < 3; ++s) {
            v2f bl = load_b_frag(Wl, s, half, col);
            vacc[t] = wmma_f32_k4(a[s], bl, vacc[t]);
        }
    }
    int seg8[8]; bool rv[8]; float sw8[8];
#pragma unroll
    for (int i = 0; i < 8; ++i) {
        long r = rowBase + i + 8 * half;
        rv[i] = (r < N);
        long rc = rv[i] ? r : (long)(N - 1);       // clamp: all loads unconditional
        int s = seg[rc];
        seg8[i] = s;
        float e = expf(weight[rc] - funmap(wmaxb[s]));
        sw8[i] = e / fmaxf(denom[s], 1e-16f);
    }
#pragma unroll
    for (int t = 0; t < 2; ++t) {
        int c = n + 16 * t;
        float b = bl_in[c];
#pragma unroll
        for (int i = 0; i < 8; ++i) {
            if (rv[i]) {
                float v = vacc[t][i] + b;
                atomicAdd(&segsum[seg8[i] * C_OUT + c], sw8[i] * v);
            }
        }
    }
}

// ---------- kernel 6: per-pillar attention MLP + final blend ----------
// second FC accumulated incrementally -> live state = mf[32] + attn[32] (no hid[64])
__global__ void k_pillar(const unsigned* __restrict__ qmax, const unsigned* __restrict__ minb,
                         const float* __restrict__ segsum, const int* __restrict__ cnt,
                         const float* __restrict__ Wfc1, const float* __restrict__ bfc1,
                         const float* __restrict__ Wfc2, const float* __restrict__ bfc2,
                         float* __restrict__ finalb, const int* __restrict__ nsegp) {
    int s = blockIdx.x * blockDim.x + threadIdx.x;
    int V = *nsegp;
    if (s >= V) return;
    float mf[C_OUT];
#pragma unroll
    for (int c = 0; c < C_OUT; ++c) {
        unsigned u = minb[s * C_OUT + c];
        mf[c] = (u == 0xFFFFFFFFu) ? 0.f : funmap(u);
    }
    float attn[C_OUT];
#pragma unroll
    for (int c = 0; c < C_OUT; ++c) attn[c] = bfc2[c];
    for (int j = 0; j < 64; ++j) {
        float acc = bfc1[j];
#pragma unroll
        for (int c = 0; c < C_OUT; ++c) acc += mf[c] * Wfc1[c * 64 + j];
        float hj = fmaxf(acc, 0.f);
#pragma unroll
        for (int c = 0; c < C_OUT; ++c) attn[c] += hj * Wfc2[j * C_OUT + c];
    }
    float cntf = fmaxf((float)cnt[s], 1.f);
#pragma unroll
    for (int c = 0; c < C_OUT; ++c) {
        float cw = 1.f / (1.f + expf(-attn[c]));
        float wx = segsum[s * C_OUT + c] / cntf;
        float qm = __uint_as_float(qmax[s * C_OUT + c]);
        finalb[s * C_OUT + c] = cw * qm + (1.f - cw) * wx;
    }
}

// ---------- kernel 7: gather final_feat back to every row ----------
__global__ void k_gather(const int* __restrict__ seg, const float* __restrict__ finalb,
                         float* __restrict__ out, int N) {
    long r = (long)blockIdx.x * blockDim.x + threadIdx.x;
    if (r >= N) return;
    int s = seg[r];
    const float4* src = (const float4*)(finalb + s * C_OUT);
    float4* dst = (float4*)(out + r * 64 + 32);
#pragma unroll
    for (int k = 0; k < 8; ++k) dst[k] = src[k];
}

// ---------- launcher ----------
extern "C" void kernel_launch(void* const* d_in, const int* in_sizes, int n_in,
                              void* d_out, int out_size, void* d_ws, size_t ws_size,
                              hipStream_t stream) {
    const float* inputs = (const float*)d_in[0];
    const int*   unq    = (const int*)d_in[1];
    const int*   nsegp  = (const int*)d_in[2];
    const float* W_lin  = (const float*)d_in[3];
    const float* b_lin  = (const float*)d_in[4];
    const float* g_norm = (const float*)d_in[5];
    const float* be_norm= (const float*)d_in[6];
    const float* W_w1   = (const float*)d_in[7];
    // d_in[8] = b_w1 (zeros; cancelled by BN mean subtraction -> folded into shift)
    const float* g_wbn  = (const float*)d_in[9];
    const float* be_wbn = (const float*)d_in[10];
    const float* W_w2   = (const float*)d_in[11];
    const float* b_w2   = (const float*)d_in[12];
    const float* W_fc1  = (const float*)d_in[13];
    const float* b_fc1  = (const float*)d_in[14];
    const float* W_fc2  = (const float*)d_in[15];
    const float* b_fc2  = (const float*)d_in[16];
    float* out = (float*)d_out;

    const int N = in_sizes[0] / IN_CH;

    // workspace layout
    char* ws = (char*)d_ws;
    size_t o = 0;
    float*    stats  = (float*)(ws + o);            o = 4096;                 // 65 f (padded)
    float*    params = (float*)(ws + 512);                                    // 128 f
    unsigned* qmax   = (unsigned*)(ws + o);         o += (size_t)VMAX * C_OUT * 4;
    unsigned* minb   = (unsigned*)(ws + o);         o += (size_t)VMAX * C_OUT * 4;
    float*    segsum = (float*)(ws + o);            o += (size_t)VMAX * C_OUT * 4;
    float*    finalb = (float*)(ws + o);            o += (size_t)VMAX * C_OUT * 4;
    unsigned* wmaxb  = (unsigned*)(ws + o);         o += (size_t)VMAX * 4;
    float*    denom  = (float*)(ws + o);            o += (size_t)VMAX * 4;
    int*      cnt    = (int*)(ws + o);              o += (size_t)VMAX * 4;
    float*    weight = (float*)(ws + o);            o += (size_t)N * 4;
    (void)ws_size; (void)n_in; (void)out_size;

    k_init<<<(VMAX * C_OUT + 255) / 256, 256, 0, stream>>>(qmax, minb, segsum, wmaxb,
                                                           denom, cnt, stats);
    k_stats<<<1024, 256, 0, stream>>>(inputs, stats, N);
    k_bnparams<<<1, 64, 0, stream>>>(stats, W_lin, g_norm, be_norm,
                                     W_w1, g_wbn, be_wbn, params, N);
    int mainBlocks = (N + 127) / 128;   // 256 threads = 8 waves = 128 rows per block
    k_main<<<mainBlocks, 256, 0, stream>>>(inputs, unq, W_lin, W_w1, W_w2, b_w2, params,
                                           out, qmax, minb, wmaxb, weight, N);
    k_denom<<<(N + 255) / 256, 256, 0, stream>>>(unq, weight, wmaxb, denom, cnt, N);
    k_wsum<<<mainBlocks, 256, 0, stream>>>(inputs, unq, W_lin, b_lin, weight, wmaxb,
                                           denom, segsum, N);
    k_pillar<<<(VMAX + 127) / 128, 128, 0, stream>>>(qmax, minb, segsum, cnt,
                                                     W_fc1, b_fc1, W_fc2, b_fc2,
                                                     finalb, nsegp);
    k_gather<<<(N + 255) / 256, 256, 0, stream>>>(unq, finalb, out, N);
}